// ROIAlign_1597727834172
// MI455X (gfx1250) — compile-verified
//
#include <hip/hip_runtime.h>
#include <hip/hip_bf16.h>

// ROI-Align on gfx1250 via per-ROI block-diagonal GEMM on the WMMA units.
// out(49x256) = A(49x784) x V(784x256) per ROI, tiled 16x16x4 f32 WMMA.
// A weights live in registers (per-lane preload); B streamed from LDS with
// one-step software pipelining so ds loads overlap v_wmma issue.

typedef __attribute__((ext_vector_type(2))) float v2f;
typedef __attribute__((ext_vector_type(8))) float v8f;

#define POOLW   7
#define NBINS   49
#define SAMP    2
#define RSCALE  0.125f
#define FC      256
#define FH      200
#define FW      304
#define FPLANE  (FH * FW)
#define CCHUNK  128
#define VSTRIDE (CCHUNK + 4)   // LDS row pad: dodge bank conflicts

__global__ __launch_bounds__(256) void roi_align_wmma_kernel(
    const float* __restrict__ feat,
    const float* __restrict__ rois,
    const float* __restrict__ roibatch,
    float* __restrict__ out)
{
    extern __shared__ float smemV[];      // [256 pixels][VSTRIDE] per (mtile,cchunk)
    __shared__ float sW[64 * 16];         // weights, bins padded to 64
    __shared__ int   sOff[64 * 16];       // flat pixel offsets within one channel plane

    const int n   = blockIdx.x;
    const int tid = threadIdx.x;

    // ---- ROI geometry (wave-uniform scalar loads) ----
    const float x1 = rois[n * 4 + 0] * RSCALE;
    const float y1 = rois[n * 4 + 1] * RSCALE;
    const float x2 = rois[n * 4 + 2] * RSCALE;
    const float y2 = rois[n * 4 + 3] * RSCALE;
    const float roi_w = fmaxf(x2 - x1, 1.0f);
    const float roi_h = fmaxf(y2 - y1, 1.0f);
    const float bin_w = roi_w / (float)POOLW;
    const float bin_h = roi_h / (float)POOLW;
    const int   bimg  = (int)roibatch[n];
    const float* fbase = feat + (size_t)bimg * FC * FPLANE;

    // ---- stage 784 (weight, pixel-offset) pairs; pad bins 49..63 with zeros ----
    for (int j = tid; j < 64 * 16; j += 256) {
        float w = 0.0f;
        int  off = 0;
        const int bin = j >> 4;
        if (bin < NBINS) {
            const int t      = j & 15;
            const int s      = t >> 2;      // sample 0..3
            const int corner = t & 3;       // dy = bit1, dx = bit0
            const int ph = bin / POOLW, pw = bin % POOLW;
            const int sy = s >> 1, sx = s & 1;
            const float gy = ((float)(ph * SAMP + sy) + 0.5f) / (float)SAMP;
            const float gx = ((float)(pw * SAMP + sx) + 0.5f) / (float)SAMP;
            const float yy = y1 + gy * bin_h;
            const float xx = x1 + gx * bin_w;
            const bool valid = (yy > -1.0f) && (yy < (float)FH) &&
                               (xx > -1.0f) && (xx < (float)FW);
            const float yc = fminf(fmaxf(yy, 0.0f), (float)(FH - 1));
            const float xc = fminf(fmaxf(xx, 0.0f), (float)(FW - 1));
            int y0 = (int)floorf(yc); y0 = min(max(y0, 0), FH - 2);
            int x0 = (int)floorf(xc); x0 = min(max(x0, 0), FW - 2);
            const float ly = yc - (float)y0;
            const float lx = xc - (float)x0;
            const int dy = (corner >> 1) & 1;
            const int dx = corner & 1;
            const float wy = dy ? ly : (1.0f - ly);
            const float wx = dx ? lx : (1.0f - lx);
            w   = valid ? (wy * wx * 0.25f) : 0.0f;   // 0.25 = mean over 2x2 samples
            off = (y0 + dy) * FW + (x0 + dx);
        }
        sW[j]   = w;
        sOff[j] = off;
    }
    __syncthreads();

    const int wave    = tid >> 5;          // wave32: 8 waves
    const int lane    = tid & 31;
    const int l15     = lane & 15;
    const int colhalf = lane >> 4;         // 0: lanes 0-15, 1: lanes 16-31
    const int c0      = wave * 16;

    for (int mt = 0; mt < 4; ++mt) {                   // 16-bin M-tiles (last padded)
        // ---- per-lane A-weight preload: this lane's bin row, its lane-half ----
        // A(16x4) layout: VGPR0 = K{2h, +0}, VGPR1 = K{2h, +1}; only row kk>>2
        // is nonzero in step kk, and lane l15 hits it for kk in [4*l15, 4*l15+4).
        v2f wq[4];
        {
            const float* wrow = &sW[(mt * 16 + l15) * 16 + (colhalf << 1)];
            #pragma unroll
            for (int q = 0; q < 4; ++q)
                wq[q] = *(const v2f*)(wrow + 4 * q);   // 8B-aligned -> ds_load_b64
        }

        for (int cc = 0; cc < FC; cc += CCHUNK) {      // 128-channel chunks
            __syncthreads();
            // ---- gather-stage V[pixel 0..255][ch 0..127] into LDS ----
            {
                const int p = tid;                     // one pixel row per thread
                const int off = sOff[mt * 256 + p];
                const float* src = fbase + (size_t)cc * FPLANE + off;
                float* dst = smemV + p * VSTRIDE;
                #pragma unroll 8
                for (int c = 0; c < CCHUNK; ++c)
                    dst[c] = src[(size_t)c * FPLANE];
            }
            __syncthreads();

            // ---- WMMA: wave handles N-tile = channels [cc + 16*wave, +16) ----
            // B(4x16) layout: lane = column N, halves split K (rows 2h, 2h+1).
            v8f acc = {};
            const float* bcol = smemV + (colhalf << 1) * VSTRIDE + c0 + l15;

            v2f Bv;                                    // pipelined B operand
            Bv.x = bcol[0];
            Bv.y = bcol[VSTRIDE];
            #pragma unroll
            for (int kk = 0; kk < 64; ++kk) {
                v2f Bn = Bv;
                if (kk < 63) {                         // prefetch next step's B
                    const float* p = bcol + (kk + 1) * 4 * VSTRIDE;
                    Bn.x = p[0];
                    Bn.y = p[VSTRIDE];
                }
                const bool act = (l15 == (kk >> 2));   // block-diagonal A
                v2f A;
                A.x = act ? wq[kk & 3].x : 0.0f;       // register cndmask, no LDS
                A.y = act ? wq[kk & 3].y : 0.0f;
                acc = __builtin_amdgcn_wmma_f32_16x16x4_f32(
                        false, A, false, Bv, (short)0, acc, false, false);
                Bv = Bn;
            }

            // ---- scatter D (16x16 f32): VGPR r -> M = r + 8*colhalf, N = l15 ----
            const int nch = cc + c0 + l15;
            float* obase = out + ((size_t)n * FC + nch) * NBINS;
            #pragma unroll
            for (int r = 0; r < 8; ++r) {
                const int bin = mt * 16 + r + colhalf * 8;
                if (bin < NBINS)
                    __builtin_nontemporal_store(acc[r], &obase[bin]);
            }
        }
    }
}

extern "C" void kernel_launch(void* const* d_in, const int* in_sizes, int n_in,
                              void* d_out, int out_size, void* d_ws, size_t ws_size,
                              hipStream_t stream) {
    const float* feat = (const float*)d_in[0];
    const float* rois = (const float*)d_in[1];
    const float* rb   = (const float*)d_in[2];
    float* out = (float*)d_out;
    const int N = in_sizes[1] / 4;                    // 512 ROIs
    const size_t shmem = (size_t)256 * VSTRIDE * sizeof(float);  // 132 KB (<= 320 KB/WGP)
    roi_align_wmma_kernel<<<N, 256, shmem, stream>>>(feat, rois, rb, out);
}